// DividedSsimLoss_11751030522115
// MI455X (gfx1250) — compile-verified
//
#include <hip/hip_runtime.h>

// ---------------------------------------------------------------------------
// Quadtree SSIM loss, MI455X (gfx1250, wave32).
// Single streaming pass (memory-bound: ~201MB @ 23.3TB/s ~ 8.6us floor).
// 512 blocks x 256 threads; thread owns an 8x8 pixel patch; block owns a
// 128x128 quadrant of a 256x256 tile. Deterministic fixed-order reductions.
// ---------------------------------------------------------------------------

typedef __attribute__((ext_vector_type(2))) float v2f;
typedef __attribute__((ext_vector_type(8))) float v8f;

#define SSIM_C1 0.2f

__device__ __forceinline__ float ssim_f(float s, float t) {
  // (2st + C1) / (s^2 + t^2 + C1)
  return fmaf(2.0f * s, t, SSIM_C1) / fmaf(s, s, fmaf(t, t, SSIM_C1));
}

// Sum the two half-wave partials per level and broadcast: lane layout has
// lane l holding partial for level (l & 15), group (l >> 4).
// WMMA f32 16x16x4: A = ones(16x4), B rows k=0 (lanes 0-15) and k=2
// (lanes 16-31) carry the partials, rows 1,3 = 0. D[m][n] = p(n) + p(n+16).
// Exact f32 arithmetic; one matrix op replaces the cross-half shuffle.
__device__ __forceinline__ float wave_levelsum(float p) {
#if __has_builtin(__builtin_amdgcn_wmma_f32_16x16x4_f32)
  v2f a; a[0] = 1.0f; a[1] = 1.0f;   // A all-ones (every K row)
  v2f b; b[0] = p;    b[1] = 0.0f;   // B rows 0,2 = partials; rows 1,3 = 0
  v8f c = {0.f, 0.f, 0.f, 0.f, 0.f, 0.f, 0.f, 0.f};
  v8f d = __builtin_amdgcn_wmma_f32_16x16x4_f32(false, a, false, b,
                                                (short)0, c, false, false);
  return d[0];  // lane n: column-n sum = total for level (lane & 15)
#else
  return p + __shfl_xor(p, 16, 32);
#endif
}

// One 2x2 sum-pool stage (n x n -> m x m) over LDS, wave-0 only.
// Accumulates the SSIM sum of the n x n *source* level while pooling.
__device__ __forceinline__ float pool_stage(const float* ss, const float* tt,
                                            float* ds, float* dt,
                                            const int n, const int lane) {
  float a = 0.0f;
  const int m = n >> 1;
  for (int p = lane; p < m * m; p += 32) {
    const int py = p / m, px = p - py * m;
    const int i00 = (2 * py) * n + 2 * px;
    const int i10 = i00 + n;
    const float s00 = ss[i00], s01 = ss[i00 + 1];
    const float s10 = ss[i10], s11 = ss[i10 + 1];
    const float u00 = tt[i00], u01 = tt[i00 + 1];
    const float u10 = tt[i10], u11 = tt[i10 + 1];
    a += ssim_f(s00, u00) + ssim_f(s01, u01) + ssim_f(s10, u10) + ssim_f(s11, u11);
    ds[p] = (s00 + s01) + (s10 + s11);
    dt[p] = (u00 + u01) + (u10 + u11);
  }
  asm volatile("s_wait_dscnt 0x0" ::: "memory");  // same-wave LDS RAW fence
  return a;
}

__global__ void __launch_bounds__(256)
ssim_main_kernel(const float* __restrict__ in, const float* __restrict__ tg,
                 float* __restrict__ wsP, float* __restrict__ wsTot) {
  __shared__ float partbuf[9][256];                 // per-level partials
  __shared__ float bufAs[256], bufAt[256];          // pyramid ping
  __shared__ float bufBs[64],  bufBt[64];           // pyramid pong

  const int t   = threadIdx.x;       // 0..255
  const int bid = blockIdx.x;        // 0..511 (tile*4 + quadrant)
  const int gi  = bid >> 2;          // global tile 0..127
  const int b   = gi >> 4;           // batch
  const int tile = gi & 15;          // 4x4 tiles per image
  const int q   = bid & 3;           // quadrant within tile
  const int row0 = (tile >> 2) * 256 + (q >> 1) * 128 + (t >> 4) * 8;
  const int col0 = (tile & 3)  * 256 + (q & 1)  * 128 + (t & 15) * 8;

  const size_t M = 1048576;          // 1024*1024 channel stride
  const size_t base = (size_t)b * 3 * M + (size_t)row0 * 1024 + col0;
  const float* iR = in + base; const float* iG = iR + M; const float* iB = iG + M;
  const float* tR = tg + base; const float* tG = tR + M; const float* tB = tG + M;

  // ---- per-thread 8x8 patch: levels 8,7,6 in registers; emit level-5 value
  float acc8 = 0.f, acc7 = 0.f, acc6 = 0.f, sT = 0.f, tT = 0.f;
  float s6r[2] = {0.f, 0.f}, t6r[2] = {0.f, 0.f};

#pragma unroll
  for (int ry = 0; ry < 4; ++ry) {              // row-pairs of pixels
    float xi[2][8], yi[2][8];
#pragma unroll
    for (int rr = 0; rr < 2; ++rr) {
      const size_t off = (size_t)(2 * ry + rr) * 1024;
#pragma unroll
      for (int k = 0; k < 2; ++k) {
        const float4 cr = *(const float4*)(iR + off + 4 * k);
        const float4 cg = *(const float4*)(iG + off + 4 * k);
        const float4 cb = *(const float4*)(iB + off + 4 * k);
        xi[rr][4*k+0] = fmaf(0.299f, cr.x, fmaf(0.587f, cg.x, 0.114f * cb.x));
        xi[rr][4*k+1] = fmaf(0.299f, cr.y, fmaf(0.587f, cg.y, 0.114f * cb.y));
        xi[rr][4*k+2] = fmaf(0.299f, cr.z, fmaf(0.587f, cg.z, 0.114f * cb.z));
        xi[rr][4*k+3] = fmaf(0.299f, cr.w, fmaf(0.587f, cg.w, 0.114f * cb.w));
        const float4 dr = *(const float4*)(tR + off + 4 * k);
        const float4 dg = *(const float4*)(tG + off + 4 * k);
        const float4 db = *(const float4*)(tB + off + 4 * k);
        yi[rr][4*k+0] = fmaf(0.299f, dr.x, fmaf(0.587f, dg.x, 0.114f * db.x));
        yi[rr][4*k+1] = fmaf(0.299f, dr.y, fmaf(0.587f, dg.y, 0.114f * db.y));
        yi[rr][4*k+2] = fmaf(0.299f, dr.z, fmaf(0.587f, dg.z, 0.114f * db.z));
        yi[rr][4*k+3] = fmaf(0.299f, dr.w, fmaf(0.587f, dg.w, 0.114f * db.w));
      }
    }
#pragma unroll
    for (int j = 0; j < 4; ++j) {               // level-7 columns
      const int c = 2 * j;
      const float x00 = xi[0][c], x01 = xi[0][c+1], x10 = xi[1][c], x11 = xi[1][c+1];
      const float y00 = yi[0][c], y01 = yi[0][c+1], y10 = yi[1][c], y11 = yi[1][c+1];
      acc8 += ssim_f(x00, y00) + ssim_f(x01, y01) + ssim_f(x10, y10) + ssim_f(x11, y11);
      const float s7 = (x00 + x01) + (x10 + x11);
      const float t7 = (y00 + y01) + (y10 + y11);
      acc7 += ssim_f(s7, t7);
      if ((ry & 1) == 0) {                      // compile-time after unroll
        if ((j & 1) == 0) { s6r[j >> 1] = s7;  t6r[j >> 1] = t7; }
        else              { s6r[j >> 1] += s7; t6r[j >> 1] += t7; }
      } else {
        s6r[j >> 1] += s7; t6r[j >> 1] += t7;
        if (j & 1) {
          const float s6 = s6r[j >> 1], t6 = t6r[j >> 1];
          acc6 += ssim_f(s6, t6);
          sT += s6; tT += t6;                   // level-5 value (patch total)
        }
      }
    }
  }

  // ---- stash per-thread results
#pragma unroll
  for (int d = 0; d < 6; ++d) partbuf[d][t] = 0.0f;
  partbuf[8][t] = acc8;
  partbuf[7][t] = acc7;
  partbuf[6][t] = acc6;
  bufAs[t] = sT;              // 16x16 level-5 grid of this quadrant (row-major)
  bufAt[t] = tT;
  __syncthreads();

  // ---- wave 0: fold levels 5..1 and produce per-level totals via WMMA
  if (t < 32) {
    const int lane = t;
    const float a5 = pool_stage(bufAs, bufAt, bufBs, bufBt, 16, lane);
    const float a4 = pool_stage(bufBs, bufBt, bufAs, bufAt,  8, lane);
    const float a3 = pool_stage(bufAs, bufAt, bufBs, bufBt,  4, lane);
    const float a2 = pool_stage(bufBs, bufBt, bufAs, bufAt,  2, lane);
    float a1 = 0.0f;
    if (lane == 0) {
      const float sQ = bufAs[0], tQ = bufAt[0];   // quadrant total = level-1 elem
      a1 = ssim_f(sQ, tQ);
      wsTot[2 * bid + 0] = sQ;
      wsTot[2 * bid + 1] = tQ;
    }
    partbuf[5][lane] = a5;
    partbuf[4][lane] = a4;
    partbuf[3][lane] = a3;
    partbuf[2][lane] = a2;
    partbuf[1][lane] = a1;
    asm volatile("s_wait_dscnt 0x0" ::: "memory");

    // lane l accumulates level (l&15), half (l>>4): 128 entries each
    const int d = lane & 15, g = lane >> 4;
    float p = 0.0f;
    if (d < 9) {
      const float* row = &partbuf[d][g * 128];
#pragma unroll 4
      for (int i = 0; i < 128; ++i) p += row[i];
    }
    const float total = wave_levelsum(p);   // EXEC all-ones here (full wave)
    if (lane < 16) wsP[bid * 16 + lane] = total;   // slot d = level-d sum
  }
}

__global__ void __launch_bounds__(256)
ssim_final_kernel(const float* __restrict__ wsP, const float* __restrict__ wsTot,
                  float* __restrict__ out) {
  __shared__ float partbuf[9][256];
  __shared__ float red[32];

  const int t = threadIdx.x;
  float local[9];
#pragma unroll
  for (int d = 0; d < 9; ++d) local[d] = 0.0f;

  // levels 8..1: fixed-order sum over the 512 block partials
  for (int i = t; i < 512; i += 256) {
#pragma unroll
    for (int d = 1; d <= 8; ++d) local[d] += wsP[i * 16 + d];
  }
  // level 0: tile i's four quadrant totals -> tile sum -> ssim
  if (t < 128) {
    const float* p = wsTot + 8 * t;
    const float s0 = (p[0] + p[2]) + (p[4] + p[6]);
    const float t0 = (p[1] + p[3]) + (p[5] + p[7]);
    local[0] = ssim_f(s0, t0);
  }
#pragma unroll
  for (int d = 0; d < 9; ++d) partbuf[d][t] = local[d];
  __syncthreads();

  if (t < 32) {
    const int d = t & 15, g = t >> 4;
    float p = 0.0f;
    if (d < 9) {
      const float* row = &partbuf[d][g * 128];
#pragma unroll 4
      for (int i = 0; i < 128; ++i) p += row[i];
    }
    const float total = wave_levelsum(p);   // S_d broadcast, full wave active
    float v = 0.0f;
    if (t < 9) {
      const float Nd = (float)(128u << (2 * t));     // 128 * 4^d elements
      v = (float)(9 - t) * (1.0f - total / Nd);      // K_LOSS[d] = 9 - d
    }
    red[t] = v;
  }
  __syncthreads();

  if (t == 0) {
    float L = 0.0f;
#pragma unroll
    for (int i = 0; i < 9; ++i) L += red[i];
    out[0] = L;
  }
}

extern "C" void kernel_launch(void* const* d_in, const int* in_sizes, int n_in,
                              void* d_out, int out_size, void* d_ws, size_t ws_size,
                              hipStream_t stream) {
  (void)in_sizes; (void)n_in; (void)out_size; (void)ws_size;
  const float* input  = (const float*)d_in[0];
  const float* target = (const float*)d_in[1];
  float* out   = (float*)d_out;
  float* wsP   = (float*)d_ws;          // 512 * 16 floats: per-block level sums
  float* wsTot = wsP + 512 * 16;        // 512 * 2 floats: quadrant (s,t) totals

  ssim_main_kernel<<<512, 256, 0, stream>>>(input, target, wsP, wsTot);
  ssim_final_kernel<<<1, 256, 0, stream>>>(wsP, wsTot, out);
}